// Net_13030930776402
// MI455X (gfx1250) — compile-verified
//
#include <hip/hip_runtime.h>
#include <cstdint>
#include <cstddef>

// Problem constants (from reference)
#define NN  38400   // NB*NPG nodes
#define EE  307200  // NB*EPG edges
#define KP1 18432   // NB*6*6 clusters after pool1
#define KP2 12800   // NB*5*5 clusters after pool2
#define KP3 2048    // NB*2*2 clusters after pool3
#define HTS (1 << 20) // hash table slots for edge dedup (>> E)

typedef __attribute__((ext_vector_type(2))) float v2f;
typedef __attribute__((ext_vector_type(8))) float v8f;

// ---------- helpers ----------
__device__ __forceinline__ unsigned enc_f(float f) {
  unsigned b = __float_as_uint(f);
  return (b & 0x80000000u) ? ~b : (b | 0x80000000u);
}
__device__ __forceinline__ float dec_f(unsigned u) {
  unsigned b = (u & 0x80000000u) ? (u & 0x7FFFFFFFu) : ~u;
  return __uint_as_float(b);
}
__device__ __forceinline__ float elu_f(float v) {
  return v > 0.0f ? v : (expf(v) - 1.0f);
}

__global__ void fill_f32(float* p, float v, int n) {
  int i = blockIdx.x * blockDim.x + threadIdx.x;
  if (i < n) p[i] = v;
}
__global__ void fill_u32(unsigned* p, unsigned v, int n) {
  int i = blockIdx.x * blockDim.x + threadIdx.x;
  if (i < n) p[i] = v;
}
__global__ void pad_w1(const float* __restrict__ W1, float* __restrict__ Wp) {
  int i = blockIdx.x * blockDim.x + threadIdx.x; // 28*32 threads
  if (i >= 28 * 32) return;
  int k = i >> 5;
  Wp[i] = (k < 25) ? W1[i] : 0.0f;
}

// ---------- spline basis: per-edge, store 4 (idx, wgt) pairs, count edges per col ----------
__global__ void basis_kernel(const float* __restrict__ src, const unsigned* __restrict__ mbits,
                             const int* __restrict__ ev, const int* __restrict__ colv,
                             float* __restrict__ cntN, int* __restrict__ idx4,
                             float* __restrict__ wgt4, int E, int mode) {
  int e = blockIdx.x * blockDim.x + threadIdx.x;
  if (e >= E) return;
  float valid = ev ? (ev[e] ? 1.0f : 0.0f) : 1.0f;
  float p0 = src[2 * e], p1 = src[2 * e + 1];
  if (mode == 1) { // src is cart; pseudo = cart/(2*m) + 0.5
    float m = fmaxf(__uint_as_float(*mbits), 1e-12f);
    float s = 1.0f / (2.0f * m);
    p0 = p0 * s + 0.5f;
    p1 = p1 * s + 0.5f;
  }
  float v0 = fminf(fmaxf(p0, 0.0f), 1.0f) * 4.0f;
  float v1 = fminf(fmaxf(p1, 0.0f), 1.0f) * 4.0f;
  float l0f = floorf(v0), l1f = floorf(v1);
  float f0 = v0 - l0f, f1 = v1 - l1f;
  int lo0 = (int)l0f, lo1 = (int)l1f;
  int hi0 = min(lo0 + 1, 4), hi1 = min(lo1 + 1, 4);
  int   i0s[2] = {lo0, hi0};  float w0s[2] = {1.0f - f0, f0};
  int   i1s[2] = {lo1, hi1};  float w1s[2] = {1.0f - f1, f1};
  int s = 0;
  #pragma unroll
  for (int m1 = 0; m1 < 2; ++m1)
    #pragma unroll
    for (int m0 = 0; m0 < 2; ++m0) {
      idx4[4 * e + s] = i0s[m0] + 5 * i1s[m1];
      wgt4[4 * e + s] = w0s[m0] * w1s[m1] * valid;
      ++s;
    }
  if (valid > 0.0f) atomicAdd(&cntN[colv[e]], 1.0f);
}

// ---------- scatter weighted source features into H[col, slot, chan] ----------
__global__ void scatter1(const float* __restrict__ x, const int* __restrict__ row,
                         const int* __restrict__ col, const int* __restrict__ idx4,
                         const float* __restrict__ wgt4, float* __restrict__ H, int E) {
  int e = blockIdx.x * blockDim.x + threadIdx.x;
  if (e >= E) return;
  float xv = x[row[e]];
  int base = col[e] * 28;
  #pragma unroll
  for (int s = 0; s < 4; ++s) {
    float w = wgt4[4 * e + s];
    if (w != 0.0f) atomicAdd(&H[base + idx4[4 * e + s]], w * xv);
  }
}

__global__ void scatterC(const float* __restrict__ x, const int* __restrict__ row,
                         const int* __restrict__ col, const int* __restrict__ idx4,
                         const float* __restrict__ wgt4, float* __restrict__ H,
                         int E, int C, int stride) {
  int t = blockIdx.x * blockDim.x + threadIdx.x;
  if (t >= E * C) return;
  int e = t / C, i = t - e * C;
  float w0 = wgt4[4 * e + 0], w1 = wgt4[4 * e + 1];
  float w2 = wgt4[4 * e + 2], w3 = wgt4[4 * e + 3];
  if (w0 + w1 + w2 + w3 == 0.0f) return;
  float xv = x[row[e] * C + i];
  int base = col[e] * stride + i;
  if (w0 != 0.0f) atomicAdd(&H[base + idx4[4 * e + 0] * C], w0 * xv);
  if (w1 != 0.0f) atomicAdd(&H[base + idx4[4 * e + 1] * C], w1 * xv);
  if (w2 != 0.0f) atomicAdd(&H[base + idx4[4 * e + 2] * C], w2 * xv);
  if (w3 != 0.0f) atomicAdd(&H[base + idx4[4 * e + 3] * C], w3 * xv);
}

// ---------- dense fp32 GEMM with V_WMMA_F32_16X16X4_F32 ----------
// C = A[M,K] @ B[K, TN*16]. One wave per 16-row strip of A; the A-fragment is
// loaded once per k-step and reused across TN independent accumulators
// (TN*16 = full output width), giving TN back-to-back WMMAs per k-step and
// 1x (not TN x) streaming of the large A matrix. M%16==0, K%4==0.
template <int TN>
__global__ void gemm_wmma(const float* __restrict__ A, const float* __restrict__ B,
                          float* __restrict__ C, int M, int K) {
  const int Nc = TN * 16;
  int wave = (int)((blockIdx.x * blockDim.x + threadIdx.x) >> 5);
  int strips = M >> 4;
  if (wave >= strips) return; // wave-uniform
  int m0 = wave << 4;
  int lane = threadIdx.x & 31;
  int half = lane >> 4;   // 0: K pair {k,k+1}; 1: {k+2,k+3}
  int l    = lane & 15;
  v8f acc[TN];
  #pragma unroll
  for (int t = 0; t < TN; ++t) acc[t] = (v8f){};
  const float* arow = A + (size_t)(m0 + l) * K + 2 * half; // row M=l, cols k+2*half..+1
  const float* bcol = B + l;                               // col N=l (+16*t), rows k+2*half..+1
  for (int k = 0; k < K; k += 4) {
    v2f a;
    a.x = arow[k];
    a.y = arow[k + 1];
    const float* bk = bcol + (size_t)(k + 2 * half) * Nc;
    #pragma unroll
    for (int t = 0; t < TN; ++t) {
      v2f b;
      b.x = bk[16 * t];
      b.y = bk[16 * t + Nc];
      acc[t] = __builtin_amdgcn_wmma_f32_16x16x4_f32(false, a, false, b, (short)0, acc[t],
                                                     false, false);
    }
  }
  #pragma unroll
  for (int t = 0; t < TN; ++t) {
    float* crow = C + (size_t)(m0 + 8 * half) * Nc + 16 * t + l;
    #pragma unroll
    for (int j = 0; j < 8; ++j) crow[(size_t)j * Nc] = acc[t][j];
  }
}

// ---------- mean-divide + root weight + bias + ELU ----------
__global__ void finalize_node(const float* __restrict__ agg, const float* __restrict__ cnt,
                              const float* __restrict__ x, const float* __restrict__ R,
                              const float* __restrict__ b, float* __restrict__ h,
                              int nn, int Cin, int Cout) {
  int t = blockIdx.x * blockDim.x + threadIdx.x;
  if (t >= nn * Cout) return;
  int n = t / Cout, o = t - n * Cout;
  float inv = 1.0f / fmaxf(cnt[n], 1.0f);
  float v = agg[t] * inv;
  for (int i = 0; i < Cin; ++i) v += x[n * Cin + i] * R[i * Cout + o];
  v += b[o];
  h[t] = elu_f(v);
}

// ---------- voxel pool: per-(node,channel) scatter ----------
__global__ void pool_scan(const float* __restrict__ x, const float* __restrict__ pos,
                          const int* __restrict__ nv, int nn, int C, float size, int grid,
                          int per_graph, int K, unsigned* __restrict__ xmax,
                          float* __restrict__ psum, float* __restrict__ cntK,
                          int* __restrict__ cl) {
  int t = blockIdx.x * blockDim.x + threadIdx.x;
  if (t >= nn * C) return;
  int n = t / C, c = t - n * C;
  int valid = nv ? nv[n] : 1;
  int cli = K;
  if (valid) {
    float px = pos[2 * n], py = pos[2 * n + 1];
    int cx = min(max((int)floorf(px / size), 0), grid - 1);
    int cy = min(max((int)floorf(py / size), 0), grid - 1);
    cli = (n / per_graph) * grid * grid + cy * grid + cx;
  }
  if (c == 0) {
    cl[n] = cli;
    if (valid) {
      atomicAdd(&psum[2 * cli], pos[2 * n]);
      atomicAdd(&psum[2 * cli + 1], pos[2 * n + 1]);
      atomicAdd(&cntK[cli], 1.0f);
    }
  }
  if (valid) atomicMax(&xmax[cli * C + c], enc_f(x[n * C + c]));
}

__global__ void pool_write(const unsigned* __restrict__ xmax, const float* __restrict__ psum,
                           const float* __restrict__ cntK, float* __restrict__ newx,
                           float* __restrict__ newpos, int* __restrict__ nvout,
                           int K, int C) {
  int t = blockIdx.x * blockDim.x + threadIdx.x;
  if (t >= K * C) return;
  int k = t / C, c = t - k * C;
  float cnt = cntK[k];
  int valid = cnt > 0.0f;
  newx[t] = valid ? dec_f(xmax[t]) : 0.0f;
  if (c == 0) {
    float inv = 1.0f / fmaxf(cnt, 1.0f);
    newpos[2 * k] = psum[2 * k] * inv;
    newpos[2 * k + 1] = psum[2 * k + 1] * inv;
    nvout[k] = valid;
  }
}

// ---------- edge remap + hash dedup (replaces lexsort+first; winner irrelevant) ----------
__global__ void edge_remap(const int* __restrict__ row, const int* __restrict__ col,
                           const int* __restrict__ ev_in, const int* __restrict__ cl, int K,
                           int* __restrict__ tab, unsigned mask, int* __restrict__ r_out,
                           int* __restrict__ c_out, int* __restrict__ ev_out, int E) {
  int e = blockIdx.x * blockDim.x + threadIdx.x;
  if (e >= E) return;
  int ok = ev_in ? ev_in[e] : 1;
  int r2 = K, c2 = K;
  if (ok) { r2 = cl[row[e]]; c2 = cl[col[e]]; }
  int valid = ok && (r2 != c2) && (r2 < K) && (c2 < K);
  int keep = 0;
  if (valid) {
    int key = r2 * K + c2; // < 2^31 for all pools
    unsigned h = ((unsigned)key * 2654435761u) & mask;
    for (;;) {
      int old = atomicCAS(&tab[h], -1, key);
      if (old == -1) { keep = 1; break; }
      if (old == key) { keep = 0; break; }
      h = (h + 1) & mask;
    }
  }
  r_out[e] = keep ? r2 : 0;
  c_out[e] = keep ? c2 : 0;
  ev_out[e] = keep;
}

__global__ void edge_cart(const int* __restrict__ r, const int* __restrict__ c,
                          const int* __restrict__ ev, const float* __restrict__ pos,
                          float* __restrict__ cart, unsigned* __restrict__ mbits, int E) {
  int e = blockIdx.x * blockDim.x + threadIdx.x;
  if (e >= E) return;
  float dx = 0.0f, dy = 0.0f;
  if (ev[e]) {
    int a = r[e], b = c[e];
    dx = pos[2 * a] - pos[2 * b];
    dy = pos[2 * a + 1] - pos[2 * b + 1];
    atomicMax(mbits, __float_as_uint(fmaxf(fabsf(dx), fabsf(dy))));
  }
  cart[2 * e] = dx;
  cart[2 * e + 1] = dy;
}

// ---------- host orchestration ----------
extern "C" void kernel_launch(void* const* d_in, const int* in_sizes, int n_in,
                              void* d_out, int out_size, void* d_ws, size_t ws_size,
                              hipStream_t stream) {
  (void)in_sizes; (void)n_in; (void)out_size; (void)ws_size;
  const float* x_in = (const float*)d_in[0];
  const float* pos  = (const float*)d_in[1];
  const float* eatt = (const float*)d_in[2];
  const float* W1 = (const float*)d_in[3];
  const float* R1 = (const float*)d_in[4];
  const float* b1 = (const float*)d_in[5];
  const float* W2 = (const float*)d_in[6];
  const float* R2 = (const float*)d_in[7];
  const float* b2 = (const float*)d_in[8];
  const float* W3 = (const float*)d_in[9];
  const float* R3 = (const float*)d_in[10];
  const float* b3 = (const float*)d_in[11];
  const int* row = (const int*)d_in[13];
  const int* col = (const int*)d_in[14];

  // workspace carve-out (256B aligned)
  size_t off = 0;
  char* base = (char*)d_ws;
  auto carve = [&](size_t bytes) -> char* {
    char* p = base + off;
    off += (bytes + 255) & ~(size_t)255;
    return p;
  };
  float* h1   = (float*)carve((size_t)NN * 32 * 4);
  float* x1   = (float*)carve((size_t)KP1 * 32 * 4);
  float* p1   = (float*)carve((size_t)KP1 * 2 * 4);
  int*   nv1  = (int*)carve((size_t)KP1 * 4);
  float* h2   = (float*)carve((size_t)KP1 * 64 * 4);
  float* x2   = (float*)carve((size_t)KP2 * 64 * 4);
  float* p2   = (float*)carve((size_t)KP2 * 2 * 4);
  int*   nv2  = (int*)carve((size_t)KP2 * 4);
  float* h3   = (float*)carve((size_t)KP2 * 64 * 4);
  int*   clb  = (int*)carve((size_t)NN * 4);
  int*   rA   = (int*)carve((size_t)EE * 4);
  int*   cA   = (int*)carve((size_t)EE * 4);
  int*   evA  = (int*)carve((size_t)EE * 4);
  int*   rB   = (int*)carve((size_t)EE * 4);
  int*   cB   = (int*)carve((size_t)EE * 4);
  int*   evB  = (int*)carve((size_t)EE * 4);
  float* cart = (float*)carve((size_t)EE * 2 * 4);
  unsigned* mb = (unsigned*)carve(256);
  int*   idx4 = (int*)carve((size_t)EE * 4 * 4);
  float* wgt4 = (float*)carve((size_t)EE * 4 * 4);
  float* cntN = (float*)carve((size_t)NN * 4);
  float* Hbuf = (float*)carve((size_t)KP2 * 1600 * 4); // largest H (layer3)
  float* agg  = (float*)carve((size_t)NN * 32 * 4);    // largest agg (layer1)
  float* W1p  = (float*)carve((size_t)28 * 32 * 4);
  unsigned* xmax = (unsigned*)carve((size_t)KP2 * 64 * 4); // largest xmax
  float* psum = (float*)carve((size_t)KP1 * 2 * 4);
  float* cntK = (float*)carve((size_t)KP1 * 4);
  int*   htab = (int*)carve((size_t)HTS * 4);
  float* p3s  = (float*)carve((size_t)KP3 * 2 * 4);
  int*   nv3s = (int*)carve((size_t)KP3 * 4);

  dim3 blk(256);
  auto gr = [](long n) { return dim3((unsigned)((n + 255) / 256)); };
  const unsigned ENC_NEG_INF = 0x007FFFFFu; // enc(-inf)
  const unsigned HMASK = HTS - 1;

  // ===== Layer 1 conv (1 -> 32) on raw graph =====
  fill_f32<<<gr((long)NN), blk, 0, stream>>>(cntN, 0.0f, NN);
  fill_f32<<<gr((long)NN * 28), blk, 0, stream>>>(Hbuf, 0.0f, NN * 28);
  pad_w1<<<gr(28 * 32), blk, 0, stream>>>(W1, W1p);
  basis_kernel<<<gr(EE), blk, 0, stream>>>(eatt, nullptr, nullptr, col, cntN, idx4, wgt4, EE, 0);
  scatter1<<<gr(EE), blk, 0, stream>>>(x_in, row, col, idx4, wgt4, Hbuf, EE);
  gemm_wmma<2><<<gr((long)(NN / 16) * 32), blk, 0, stream>>>(Hbuf, W1p, agg, NN, 28);
  finalize_node<<<gr((long)NN * 32), blk, 0, stream>>>(agg, cntN, x_in, R1, b1, h1, NN, 1, 32);

  // ===== Pool 1 (size 5, grid 6) =====
  fill_u32<<<gr((long)KP1 * 32), blk, 0, stream>>>(xmax, ENC_NEG_INF, KP1 * 32);
  fill_f32<<<gr((long)KP1 * 2), blk, 0, stream>>>(psum, 0.0f, KP1 * 2);
  fill_f32<<<gr((long)KP1), blk, 0, stream>>>(cntK, 0.0f, KP1);
  pool_scan<<<gr((long)NN * 32), blk, 0, stream>>>(h1, pos, nullptr, NN, 32, 5.0f, 6, 75, KP1,
                                                   xmax, psum, cntK, clb);
  pool_write<<<gr((long)KP1 * 32), blk, 0, stream>>>(xmax, psum, cntK, x1, p1, nv1, KP1, 32);
  fill_u32<<<gr((long)HTS), blk, 0, stream>>>((unsigned*)htab, 0xFFFFFFFFu, HTS);
  edge_remap<<<gr(EE), blk, 0, stream>>>(row, col, nullptr, clb, KP1, htab, HMASK, rA, cA, evA, EE);
  fill_u32<<<gr(1), blk, 0, stream>>>(mb, 0u, 1);
  edge_cart<<<gr(EE), blk, 0, stream>>>(rA, cA, evA, p1, cart, mb, EE);

  // ===== Layer 2 conv (32 -> 64) =====
  fill_f32<<<gr((long)KP1), blk, 0, stream>>>(cntN, 0.0f, KP1);
  fill_f32<<<gr((long)KP1 * 800), blk, 0, stream>>>(Hbuf, 0.0f, KP1 * 800);
  basis_kernel<<<gr(EE), blk, 0, stream>>>(cart, mb, evA, cA, cntN, idx4, wgt4, EE, 1);
  scatterC<<<gr((long)EE * 32), blk, 0, stream>>>(x1, rA, cA, idx4, wgt4, Hbuf, EE, 32, 800);
  gemm_wmma<4><<<gr((long)(KP1 / 16) * 32), blk, 0, stream>>>(Hbuf, W2, agg, KP1, 800);
  finalize_node<<<gr((long)KP1 * 64), blk, 0, stream>>>(agg, cntN, x1, R2, b2, h2, KP1, 32, 64);

  // ===== Pool 2 (size 7, grid 5) =====
  fill_u32<<<gr((long)KP2 * 64), blk, 0, stream>>>(xmax, ENC_NEG_INF, KP2 * 64);
  fill_f32<<<gr((long)KP2 * 2), blk, 0, stream>>>(psum, 0.0f, KP2 * 2);
  fill_f32<<<gr((long)KP2), blk, 0, stream>>>(cntK, 0.0f, KP2);
  pool_scan<<<gr((long)KP1 * 64), blk, 0, stream>>>(h2, p1, nv1, KP1, 64, 7.0f, 5, 36, KP2,
                                                    xmax, psum, cntK, clb);
  pool_write<<<gr((long)KP2 * 64), blk, 0, stream>>>(xmax, psum, cntK, x2, p2, nv2, KP2, 64);
  fill_u32<<<gr((long)HTS), blk, 0, stream>>>((unsigned*)htab, 0xFFFFFFFFu, HTS);
  edge_remap<<<gr(EE), blk, 0, stream>>>(rA, cA, evA, clb, KP2, htab, HMASK, rB, cB, evB, EE);
  fill_u32<<<gr(1), blk, 0, stream>>>(mb, 0u, 1);
  edge_cart<<<gr(EE), blk, 0, stream>>>(rB, cB, evB, p2, cart, mb, EE);

  // ===== Layer 3 conv (64 -> 64) =====
  fill_f32<<<gr((long)KP2), blk, 0, stream>>>(cntN, 0.0f, KP2);
  fill_f32<<<gr((long)KP2 * 1600), blk, 0, stream>>>(Hbuf, 0.0f, KP2 * 1600);
  basis_kernel<<<gr(EE), blk, 0, stream>>>(cart, mb, evB, cB, cntN, idx4, wgt4, EE, 1);
  scatterC<<<gr((long)EE * 64), blk, 0, stream>>>(x2, rB, cB, idx4, wgt4, Hbuf, EE, 64, 1600);
  gemm_wmma<4><<<gr((long)(KP2 / 16) * 32), blk, 0, stream>>>(Hbuf, W3, agg, KP2, 1600);
  finalize_node<<<gr((long)KP2 * 64), blk, 0, stream>>>(agg, cntN, x2, R3, b3, h3, KP2, 64, 64);

  // ===== Pool 3 (size 14, grid 2) -> d_out [2048, 64] =====
  fill_u32<<<gr((long)KP3 * 64), blk, 0, stream>>>(xmax, ENC_NEG_INF, KP3 * 64);
  fill_f32<<<gr((long)KP3 * 2), blk, 0, stream>>>(psum, 0.0f, KP3 * 2);
  fill_f32<<<gr((long)KP3), blk, 0, stream>>>(cntK, 0.0f, KP3);
  pool_scan<<<gr((long)KP2 * 64), blk, 0, stream>>>(h3, p2, nv2, KP2, 64, 14.0f, 2, 25, KP3,
                                                    xmax, psum, cntK, clb);
  pool_write<<<gr((long)KP3 * 64), blk, 0, stream>>>(xmax, psum, cntK, (float*)d_out, p3s, nv3s,
                                                     KP3, 64);
}